// PolyakovLoopGenerator_20023137534661
// MI455X (gfx1250) — compile-verified
//
#include <hip/hip_runtime.h>

// Polyakov loops on a 24^4 lattice, 4 directions, 3x3 complex links (f32).
// Two-sided scan per line: even-k prefixes T_0,T_2,...,T_22 stored in LDS,
// suffix S_k streamed backwards via async global->LDS ring (ASYNCcnt path),
// odd prefixes reconstructed on the fly: T_k = T_{k-1} * U(k-1).
// P_k = S_k * T_k.  81 matmuls/line vs 552 naive; 36 KB LDS / workgroup.

#define LL        24
#define NLINES    (LL*LL*LL)      // 13824 lines per direction
#define SITES     (LL*LL*LL*LL)   // 331776
#define PLANE     (SITES*9)       // floats per direction plane = 2985984
#define RINGSLOTS 4               // LDS ring slots (3-matrix lookahead)

// ---- 3x3 complex matmul: C = A * B (C must not alias A/B) -----------------
__device__ __forceinline__ void cmm(const float* __restrict__ ar, const float* __restrict__ ai,
                                    const float* __restrict__ br, const float* __restrict__ bi,
                                    float* __restrict__ cr, float* __restrict__ ci) {
#pragma unroll
  for (int i = 0; i < 3; ++i) {
#pragma unroll
    for (int j = 0; j < 3; ++j) {
      float re = 0.f, im = 0.f;
#pragma unroll
      for (int m = 0; m < 3; ++m) {
        const float a_r = ar[i*3+m], a_i = ai[i*3+m];
        const float b_r = br[m*3+j], b_i = bi[m*3+j];
        re = fmaf(a_r, b_r, re); re = fmaf(-a_i, b_i, re);
        im = fmaf(a_r, b_i, im); im = fmaf(a_i, b_r, im);
      }
      cr[i*3+j] = re; ci[i*3+j] = im;
    }
  }
}

// ---- CDNA5 async global -> LDS (ASYNCcnt path) ----------------------------
__device__ __forceinline__ void async_ld_f32(unsigned lds_byte_addr, const float* gaddr) {
#if defined(__gfx1250__)
  asm volatile("global_load_async_to_lds_b32 %0, %1, off"
               :: "v"(lds_byte_addr), "v"(gaddr)
               : "memory");
#endif
}

// Async loads complete in order. We prime 3 matrices (54 issues) and issue
// exactly 18 per consumed matrix, so before consuming matrix m a total of
// 18*(3+m) are issued; asynccnt<=36 guarantees >=18*(m+1) complete, i.e. the
// matrix we are about to read is resident. Outstanding never exceeds 54 < 63
// (6-bit ASYNCcnt), so issue never stalls.
__device__ __forceinline__ void wait_async_le_36() {
#if defined(__gfx1250__)
  asm volatile("s_wait_asynccnt 36" ::: "memory");
#endif
}

__global__ __launch_bounds__(32)
void polyakov_lines_kernel(const float* __restrict__ U_re,
                           const float* __restrict__ U_im,
                           float* __restrict__ out) {
  // Lane-private LDS, [slot][elem][lane] layout -> bank-conflict free.
  __shared__ float sT[12 * 18 * 32];        // even-k prefixes: 27,648 B
  __shared__ float sU[RINGSLOTS * 18 * 32]; // async ring:       9,216 B (36 KB total)

  const int lane = threadIdx.x;
  const int line = blockIdx.x * 32 + lane;  // 432 blocks * 32 = 13824 lines
  const int mu   = blockIdx.y;

  int siteBase, strideS;                    // x3-fastest line order
  switch (mu) {
    case 0:  siteBase = line;                                strideS = 13824; break;
    case 1:  siteBase = (line / 576) * 13824 + (line % 576); strideS = 576;   break;
    case 2:  siteBase = (line / 24) * 576 + (line % 24);     strideS = 24;    break;
    default: siteBase = line * 24;                           strideS = 1;     break;
  }
  const long gbase = (long)mu * PLANE + (long)siteBase * 9;
  const int  sF    = strideS * 9;
  const float* __restrict__ gre = U_re + gbase;
  const float* __restrict__ gim = U_im + gbase;
  float* __restrict__ out_re = out;
  float* __restrict__ out_im = out + 4l * PLANE;

  // ---- Prime the async ring with U(23),U(22),U(21) (consume idx 0,1,2) ----
#pragma unroll
  for (int m = 0; m < 3; ++m) {
    const int kk = 23 - m;
#pragma unroll
    for (int e = 0; e < 9; ++e) {
      async_ld_f32((unsigned)(size_t)&sU[((m * 18) + e)     * 32 + lane], gre + kk * sF + e);
      async_ld_f32((unsigned)(size_t)&sU[((m * 18) + 9 + e) * 32 + lane], gim + kk * sF + e);
    }
  }

  // ---- Pass A: prefix scan; store T_k only for even k (slot k/2) ----------
  float Tr[9], Ti[9];
#pragma unroll
  for (int e = 0; e < 9; ++e) { Tr[e] = (e == 0 || e == 4 || e == 8) ? 1.f : 0.f; Ti[e] = 0.f; }
#pragma unroll
  for (int e = 0; e < 9; ++e) { sT[e * 32 + lane] = Tr[e]; sT[(9 + e) * 32 + lane] = Ti[e]; }

  float Cr[9], Ci[9];                        // current U(k), double-buffered
#pragma unroll
  for (int e = 0; e < 9; ++e) { Cr[e] = gre[e]; Ci[e] = gim[e]; }

  for (int k = 0; k < 23; ++k) {
    float Xr[9], Xi[9];                      // next U(k+1)
    if (k < 22) {
#pragma unroll
      for (int e = 0; e < 9; ++e) { Xr[e] = gre[(k + 1) * sF + e]; Xi[e] = gim[(k + 1) * sF + e]; }
    }
    int pk = k + 6; if (pk >= 24) pk -= 24;  // deep prefetch (global_prefetch_b8)
    __builtin_prefetch(gre + pk * sF, 0, 3);
    __builtin_prefetch(gim + pk * sF, 0, 3);

    float Nr[9], Ni[9];
    cmm(Tr, Ti, Cr, Ci, Nr, Ni);             // T_{k+1} = T_k * U(k)
#pragma unroll
    for (int e = 0; e < 9; ++e) { Tr[e] = Nr[e]; Ti[e] = Ni[e]; }
    if ((k & 1) == 1) {                      // k+1 even -> store slot (k+1)/2
      const int slot = (k + 1) >> 1;
#pragma unroll
      for (int e = 0; e < 9; ++e) {
        sT[(slot * 18 + e)     * 32 + lane] = Tr[e];
        sT[(slot * 18 + 9 + e) * 32 + lane] = Ti[e];
      }
    }
    if (k < 22) {
#pragma unroll
      for (int e = 0; e < 9; ++e) { Cr[e] = Xr[e]; Ci[e] = Xi[e]; }
    }
  }

  // ---- Pass B: suffix scan backwards, two sites (odd k, even k-1) per step
  float Sr[9], Si[9];
  for (int p = 0; p < 12; ++p) {
    const int ko = 23 - 2 * p;               // odd k   (consume idx 2p)
    const int ke = ko - 1;                   // even k  (consume idx 2p+1)

    // consume U(ko)
    wait_async_le_36();
    {
      const int slot = (2 * p) & 3;
      float Ur[9], Ui[9];
#pragma unroll
      for (int e = 0; e < 9; ++e) {
        Ur[e] = sU[(slot * 18 + e)     * 32 + lane];
        Ui[e] = sU[(slot * 18 + 9 + e) * 32 + lane];
      }
      int kk = 20 - 2 * p; if (kk < 0) kk += 24;     // refill (dummy at tail)
      const int ws = (2 * p + 3) & 3;
#pragma unroll
      for (int e = 0; e < 9; ++e) {
        async_ld_f32((unsigned)(size_t)&sU[((ws * 18) + e)     * 32 + lane], gre + kk * sF + e);
        async_ld_f32((unsigned)(size_t)&sU[((ws * 18) + 9 + e) * 32 + lane], gim + kk * sF + e);
      }
      if (p == 0) {
#pragma unroll
        for (int e = 0; e < 9; ++e) { Sr[e] = Ur[e]; Si[e] = Ui[e]; }   // S_23 = U(23)
      } else {
        float Nr[9], Ni[9];
        cmm(Ur, Ui, Sr, Si, Nr, Ni);                                    // S_ko = U(ko)*S_{ko+1}
#pragma unroll
        for (int e = 0; e < 9; ++e) { Sr[e] = Nr[e]; Si[e] = Ni[e]; }
      }
    }

    // consume U(ke)
    wait_async_le_36();
    float Uer[9], Uei[9];
    {
      const int slot = (2 * p + 1) & 3;
#pragma unroll
      for (int e = 0; e < 9; ++e) {
        Uer[e] = sU[(slot * 18 + e)     * 32 + lane];
        Uei[e] = sU[(slot * 18 + 9 + e) * 32 + lane];
      }
      int kk = 19 - 2 * p; if (kk < 0) kk += 24;
      const int ws = (2 * p + 4) & 3;
#pragma unroll
      for (int e = 0; e < 9; ++e) {
        async_ld_f32((unsigned)(size_t)&sU[((ws * 18) + e)     * 32 + lane], gre + kk * sF + e);
        async_ld_f32((unsigned)(size_t)&sU[((ws * 18) + 9 + e) * 32 + lane], gim + kk * sF + e);
      }
    }

    // T_ke from LDS (even slot 11-p); T_ko = T_ke * U(ke)
    float Ter[9], Tei[9];
    {
      const int slot = 11 - p;
#pragma unroll
      for (int e = 0; e < 9; ++e) {
        Ter[e] = sT[(slot * 18 + e)     * 32 + lane];
        Tei[e] = sT[(slot * 18 + 9 + e) * 32 + lane];
      }
    }
    float Tor[9], Toi[9];
    cmm(Ter, Tei, Uer, Uei, Tor, Toi);       // T_ko = T_{ko-1} * U(ko-1)

    float Pr[9], Pi[9];
    cmm(Sr, Si, Tor, Toi, Pr, Pi);           // P_ko = S_ko * T_ko
    {
      const long so = gbase + (long)ko * sF;
#pragma unroll
      for (int e = 0; e < 9; ++e) {
        __builtin_nontemporal_store(Pr[e], &out_re[so + e]);
        __builtin_nontemporal_store(Pi[e], &out_im[so + e]);
      }
    }

    {
      float Nr[9], Ni[9];
      cmm(Uer, Uei, Sr, Si, Nr, Ni);         // S_ke = U(ke) * S_ko
#pragma unroll
      for (int e = 0; e < 9; ++e) { Sr[e] = Nr[e]; Si[e] = Ni[e]; }
    }
    if (ke == 0) {
#pragma unroll
      for (int e = 0; e < 9; ++e) { Pr[e] = Sr[e]; Pi[e] = Si[e]; }     // T_0 = I
    } else {
      cmm(Sr, Si, Ter, Tei, Pr, Pi);         // P_ke = S_ke * T_ke
    }
    {
      const long so = gbase + (long)ke * sF;
#pragma unroll
      for (int e = 0; e < 9; ++e) {
        __builtin_nontemporal_store(Pr[e], &out_re[so + e]);
        __builtin_nontemporal_store(Pi[e], &out_im[so + e]);
      }
    }
  }
}

extern "C" void kernel_launch(void* const* d_in, const int* in_sizes, int n_in,
                              void* d_out, int out_size, void* d_ws, size_t ws_size,
                              hipStream_t stream) {
  (void)in_sizes; (void)n_in; (void)d_ws; (void)ws_size; (void)out_size;
  const float* U_re = (const float*)d_in[0];
  const float* U_im = (const float*)d_in[1];
  float* out = (float*)d_out;
  dim3 grid(NLINES / 32, 4, 1);  // 432 blocks x 4 directions
  dim3 block(32, 1, 1);          // one wave32 per workgroup, one line per lane
  polyakov_lines_kernel<<<grid, block, 0, stream>>>(U_re, U_im, out);
}